// SOGAE2_88330297409969
// MI455X (gfx1250) — compile-verified
//
#include <hip/hip_runtime.h>

// loss[b] = (1/G) * sum_g images[b,g]^2   (exact reduction of the reference:
// decode carry starts at x=0 -> dz=0 -> m=0 -> x_rec stays identically 0).
//
// WMMA Gram-diagonal reduction: C += A * A^T via v_wmma_f32_16x16x4_f32 with
// the same VGPR pair fed as both A and B (B=A^T has the identical register
// image for the 32-bit 16x4 operand layout). diag(C) accumulates sum(x^2)
// per row. Column->K-slot permutation inside a WMMA is irrelevant for the
// diagonal, so each lane streams float4 (b128) loads directly into the MACs.

typedef __attribute__((ext_vector_type(2))) float v2f;
typedef __attribute__((ext_vector_type(4))) float v4f;
typedef __attribute__((ext_vector_type(8))) float v8f;

#define G_DIM 8192
#define B_DIM 4096
#define ROWS_PER_BLOCK 16
#define WAVES_PER_BLOCK 8
#define THREADS_PER_BLOCK (WAVES_PER_BLOCK * 32)
#define COLS_PER_WAVE (G_DIM / WAVES_PER_BLOCK) /* 1024 */
#define K_ITERS (COLS_PER_WAVE / 8)             /* 128: 8 cols per iter */

__global__ __launch_bounds__(THREADS_PER_BLOCK, 1)
void sogae2_rowsumsq_wmma(const float* __restrict__ images,
                          float* __restrict__ out) {
    __shared__ float red[WAVES_PER_BLOCK][32][8];

    const int tid  = threadIdx.x;
    const int wave = tid >> 5;
    const int lane = tid & 31;
    const int row  = lane & 15;   // M index of the A-matrix row this lane holds
    const int half = lane >> 4;   // K half (lanes 16-31 hold K=2,3 slots)

    const int rowBase = blockIdx.x * ROWS_PER_BLOCK;
    const float* __restrict__ p = images
        + (size_t)(rowBase + row) * G_DIM
        + wave * COLS_PER_WAVE
        + half * 4;

    v8f c0 = {};
    v8f c1 = {};

#pragma unroll 4
    for (int it = 0; it < K_ITERS; ++it) {
        // lanes L and L+16 together cover 32 contiguous bytes of row L&15
        v4f d = *(const v4f*)(p + it * 8);
        v2f a0 = { d.x, d.y };
        v2f a1 = { d.z, d.w };
        // C += A * A^T : same registers as A and B operands
        c0 = __builtin_amdgcn_wmma_f32_16x16x4_f32(
                 false, a0, false, a0, (short)0, c0, false, false);
        c1 = __builtin_amdgcn_wmma_f32_16x16x4_f32(
                 false, a1, false, a1, (short)0, c1, false, false);
    }

    // spill each lane's 8 accumulator rows (diag(C0)+diag(C1) lives in here)
#pragma unroll
    for (int i = 0; i < 8; ++i)
        red[wave][lane][i] = c0[i] + c1[i];

    __syncthreads();

    // D[m][m]: m<8 -> VGPR m, lane m ; m>=8 -> VGPR m-8, lane m+16
    if (tid < ROWS_PER_BLOCK) {
        const int m       = tid;
        const int srcLane = (m < 8) ? m : (m + 16);
        const int srcReg  = (m < 8) ? m : (m - 8);
        float s = 0.0f;
#pragma unroll
        for (int w = 0; w < WAVES_PER_BLOCK; ++w)
            s += red[w][srcLane][srcReg];
        out[rowBase + m] = s * (1.0f / (float)G_DIM);
    }
}

extern "C" void kernel_launch(void* const* d_in, const int* in_sizes, int n_in,
                              void* d_out, int out_size, void* d_ws, size_t ws_size,
                              hipStream_t stream) {
    (void)in_sizes; (void)n_in; (void)d_ws; (void)ws_size; (void)out_size;
    const float* images = (const float*)d_in[0];   // (4096, 8192) f32
    // d_in[1] = xw, d_in[2] = xw2, d_in[3] = z0 : provably dead in the
    // reference computation (decode state x is identically zero).
    float* out = (float*)d_out;                    // (4096,) f32

    dim3 grid(B_DIM / ROWS_PER_BLOCK);             // 256 blocks
    dim3 block(THREADS_PER_BLOCK);                 // 8 wave32s
    sogae2_rowsumsq_wmma<<<grid, block, 0, stream>>>(images, out);
}